// ChamferLoss_60619168416150
// MI455X (gfx1250) — compile-verified
//
#include <hip/hip_runtime.h>
#include <float.h>

typedef __attribute__((ext_vector_type(2))) float v2f;
typedef __attribute__((ext_vector_type(4))) float v4f;
typedef __attribute__((ext_vector_type(8))) float v8f;

#define BATCH  8
#define NPTS   8192            // M == N == 8192
#define TOTAL  (BATCH * NPTS)

#define MPW    4               // 16-row A tiles per wave (64 rows / wave)
#define NCH    4               // column chunks per row-group
#define NCHSZ  (NPTS / NCH)    // 2048 columns per chunk
#define NTILES (NCHSZ / 16)    // 128 column tiles per wave (even)
#define MGRPS  (NPTS / (16 * MPW))          // 128 row-groups per batch
#define WAVES  (BATCH * MGRPS * NCH)        // 4096 waves per pass

// Single-instruction IEEE maximumNumber ops (inputs provably NaN-free).
__device__ __forceinline__ float vmax(float a, float b) {
    float r;
    asm("v_max_num_f32 %0, %1, %2" : "=v"(r) : "v"(a), "v"(b));
    return r;
}
__device__ __forceinline__ float vmax3(float a, float b, float c) {
    float r;
    asm("v_max3_num_f32 %0, %1, %2, %3" : "=v"(r) : "v"(a), "v"(b), "v"(c));
    return r;
}

// Workspace (floats): rowMaxT[T], rowMaxP[T], tgtPad[4T], prdPad[4T] = 2.5MB
// Each point stored as {x, y, z, -0.5*|s|^2}:
//   - as B operand: K=3 slot supplies the -0.5*|col|^2 rank-1 bias
//   - as A operand: w patched to 1.0 in registers (multiplies that bias)
//   - its .w read back as the constant C bias (-0.5*|row|^2)

// ---------------------------------------------------------------------------
__global__ __launch_bounds__(256) void chamfer_prep(const float* __restrict__ pred,
                                                    const float* __restrict__ target,
                                                    float* __restrict__ rowMaxT,
                                                    float* __restrict__ rowMaxP,
                                                    v4f* __restrict__ tgtPad,
                                                    v4f* __restrict__ prdPad) {
    int i = blockIdx.x * 256 + threadIdx.x;
    if (i >= TOTAL) return;
    float px = pred[3 * i], py = pred[3 * i + 1], pz = pred[3 * i + 2];
    float tx = target[3 * i], ty = target[3 * i + 1], tz = target[3 * i + 2];
    prdPad[i] = (v4f){px, py, pz, -0.5f * (px * px + py * py + pz * pz)};
    tgtPad[i] = (v4f){tx, ty, tz, -0.5f * (tx * tx + ty * ty + tz * tz)};
    rowMaxT[i] = -FLT_MAX;
    rowMaxP[i] = -FLT_MAX;
}

// ---------------------------------------------------------------------------
// One directed pass: for each A row, max over all B columns of
//   D = A.B - 0.5|A|^2 - 0.5|B|^2   (dist = -2*D, so min-dist == max-D).
// Two column tiles per iteration so each accumulator update is one
// v_max3_num_f32 over both tiles' results.
// ---------------------------------------------------------------------------
__global__ __launch_bounds__(256) void chamfer_pass(float* __restrict__ rowOut,
                                                    const float* __restrict__ aPad,
                                                    const float* __restrict__ bPad) {
    const int lane   = threadIdx.x & 31;
    const int wid    = threadIdx.x >> 5;
    const int waveId = blockIdx.x * 8 + wid;
    const int b      = waveId / (MGRPS * NCH);
    const int rem    = waveId % (MGRPS * NCH);
    const int mg     = rem >> 2;               // row-group (0..127)
    const int nc     = rem & (NCH - 1);        // column chunk (0..3)
    const int r      = lane & 15;
    const int half   = lane >> 4;              // selects K pair (2h, 2h+1)

    const int m0 = b * NPTS + mg * (16 * MPW);

    // A fragments (patch w -> 1.0) + constant C bias (-0.5*|row|^2)
    v2f av[MPW];
    v8f cbias[MPW];
#pragma unroll
    for (int t = 0; t < MPW; ++t) {
        av[t] = *(const v2f*)(aPad + (size_t)(m0 + t * 16 + r) * 4 + 2 * half);
        if (half) av[t].y = 1.0f;              // A[m][3] = 1 multiplies B's bias
#pragma unroll
        for (int k = 0; k < 8; ++k)
            cbias[t][k] = aPad[(size_t)(m0 + t * 16 + k + 8 * half) * 4 + 3];
    }

    float rmax[MPW][8];
#pragma unroll
    for (int t = 0; t < MPW; ++t)
#pragma unroll
        for (int k = 0; k < 8; ++k) rmax[t][k] = -FLT_MAX;

    const float* bF    = bPad + (size_t)b * NPTS * 4;
    const int    nBase = nc * NCHSZ;
    const int    fragOff = 2 * half;

    // software pipeline: prefetch the first pair of B fragments
    v2f bv0 = *(const v2f*)(bF + (size_t)(nBase + 0 * 16 + r) * 4 + fragOff);
    v2f bv1 = *(const v2f*)(bF + (size_t)(nBase + 1 * 16 + r) * 4 + fragOff);

    for (int it = 0; it < NTILES; it += 2) {
        const bool more = (it + 2) < NTILES;
        const int  i2   = more ? it + 2 : it;
        const int  i3   = more ? it + 3 : it + 1;
        v2f bv2 = *(const v2f*)(bF + (size_t)(nBase + i2 * 16 + r) * 4 + fragOff);
        v2f bv3 = *(const v2f*)(bF + (size_t)(nBase + i3 * 16 + r) * 4 + fragOff);

#pragma unroll
        for (int t = 0; t < MPW; ++t) {
            v8f da = __builtin_amdgcn_wmma_f32_16x16x4_f32(
                         false, av[t], false, bv0, (short)0, cbias[t], false, false);
            v8f db = __builtin_amdgcn_wmma_f32_16x16x4_f32(
                         false, av[t], false, bv1, (short)0, cbias[t], false, false);
#pragma unroll
            for (int k = 0; k < 8; ++k)
                rmax[t][k] = vmax3(rmax[t][k], da[k], db[k]);
        }
        bv0 = bv2;
        bv1 = bv3;
    }

    // row maxima: reduce across the 16 lanes of each half, then one-time
    // atomic (NCH chunk-waves share each row).
#pragma unroll
    for (int t = 0; t < MPW; ++t) {
#pragma unroll
        for (int k = 0; k < 8; ++k) {
            float v = rmax[t][k];
            v = vmax(v, __shfl_xor(v, 8, 32));
            v = vmax(v, __shfl_xor(v, 4, 32));
            v = vmax(v, __shfl_xor(v, 2, 32));
            v = vmax(v, __shfl_xor(v, 1, 32));
            rmax[t][k] = v;
        }
    }
    if (r == 0) {
#pragma unroll
        for (int t = 0; t < MPW; ++t)
#pragma unroll
            for (int k = 0; k < 8; ++k)
                atomicMax(&rowOut[m0 + t * 16 + k + 8 * half], rmax[t][k]);
    }
}

// ---------------------------------------------------------------------------
// Final: loss = mean(-2*rowMaxT) + mean(-2*rowMaxP).
// ---------------------------------------------------------------------------
__global__ __launch_bounds__(256) void chamfer_reduce(const float* __restrict__ rowMaxT,
                                                      const float* __restrict__ rowMaxP,
                                                      float* __restrict__ out) {
    __shared__ float s1[256], s2[256];
    float a = 0.0f, c = 0.0f;
    for (int i = threadIdx.x; i < TOTAL; i += 256) {
        a += rowMaxT[i];
        c += rowMaxP[i];
    }
    s1[threadIdx.x] = a;
    s2[threadIdx.x] = c;
    __syncthreads();
    for (int s = 128; s > 0; s >>= 1) {
        if (threadIdx.x < s) {
            s1[threadIdx.x] += s1[threadIdx.x + s];
            s2[threadIdx.x] += s2[threadIdx.x + s];
        }
        __syncthreads();
    }
    if (threadIdx.x == 0) {
        const float scale = -2.0f / (float)TOTAL;   // undo the -0.5 folding
        out[0] = (s1[0] + s2[0]) * scale;
    }
}

extern "C" void kernel_launch(void* const* d_in, const int* in_sizes, int n_in,
                              void* d_out, int out_size, void* d_ws, size_t ws_size,
                              hipStream_t stream) {
    const float* pred   = (const float*)d_in[0];
    const float* target = (const float*)d_in[1];
    float* out = (float*)d_out;

    float* ws      = (float*)d_ws;        // 10 * TOTAL floats = 2.5 MB
    float* rowMaxT = ws;                  // per-target-row max (dist1)
    float* rowMaxP = ws + TOTAL;          // per-pred-row   max (dist2)
    v4f*   tgtPad  = (v4f*)(ws + 2 * TOTAL);
    v4f*   prdPad  = (v4f*)(ws + 6 * TOTAL);

    chamfer_prep<<<TOTAL / 256, 256, 0, stream>>>(pred, target, rowMaxT, rowMaxP,
                                                  tgtPad, prdPad);
    chamfer_pass<<<WAVES / 8, 256, 0, stream>>>(rowMaxT, (const float*)tgtPad,
                                                (const float*)prdPad);
    chamfer_pass<<<WAVES / 8, 256, 0, stream>>>(rowMaxP, (const float*)prdPad,
                                                (const float*)tgtPad);
    chamfer_reduce<<<1, 256, 0, stream>>>(rowMaxT, rowMaxP, out);
}